// LinearAttention_8297876815995
// MI455X (gfx1250) — compile-verified
//
#include <hip/hip_runtime.h>

// CDNA5 (gfx1250) wave32 fp32-WMMA implementation of the no-softmax linear
// attention classifier.  (p p^T) p == p (p^T p) collapses attention to a
// 64x64 Gram matrix per head.  The two large GEMMs are LDS-staged and
// double-buffered, using GLOBAL_LOAD_ASYNC_TO_LDS_B128 (ASYNCcnt) so WMMAs
// are fed from LDS instead of stalling on per-load s_wait_loadcnt 0.

typedef __attribute__((ext_vector_type(2))) float v2f;
typedef __attribute__((ext_vector_type(4))) float v4f;
typedef __attribute__((ext_vector_type(8))) float v8f;
typedef __attribute__((ext_vector_type(4))) int   v4i;

#define BS        2
#define SEQ       2048
#define EMBED     1024
#define NHEADS    16
#define HDIM      64
#define NCLASSES  1000

// Large-GEMM tiling
#define BM 128
#define BN 64
#define BK 32
#define LDST 36          // LDS row stride (floats): 144B -> 16B aligned rows,
                         // and 36*n mod 64 is a permutation over n=0..15
                         // (conflict-free 16-lane WMMA fragment reads)

#if defined(__has_builtin)
#if __has_builtin(__builtin_amdgcn_global_load_async_to_lds_b128)
#define HAVE_ASYNC_LDS 1
#endif
#endif
#ifndef HAVE_ASYNC_LDS
#define HAVE_ASYNC_LDS 0
#endif

static __device__ __forceinline__ v8f wmma_f32(v2f a, v2f b, v8f c) {
  // 8 args: (neg_a, A, neg_b, B, c_mod, C, reuse_a, reuse_b)
  return __builtin_amdgcn_wmma_f32_16x16x4_f32(false, a, false, b, (short)0, c,
                                               false, false);
}

#if HAVE_ASYNC_LDS
static __device__ __forceinline__ void wait_async_le6() {
#if __has_builtin(__builtin_amdgcn_s_wait_asynccnt)
  __builtin_amdgcn_s_wait_asynccnt(6);
#else
  asm volatile("s_wait_asynccnt 6" ::: "memory");
#endif
}
static __device__ __forceinline__ void wait_async_le0() {
#if __has_builtin(__builtin_amdgcn_s_wait_asynccnt)
  __builtin_amdgcn_s_wait_asynccnt(0);
#else
  asm volatile("s_wait_asynccnt 0" ::: "memory");
#endif
}
#endif

// ---------------------------------------------------------------------------
// C[M,N] = A[M,K] @ Wt[N,K]^T + bias, LDS-staged, double-buffered.
// grid = (ceil(N/64), M/128), block = 256 (8 waves).
// Wave w computes rows [m0+16w, m0+16w+16) x all 64 N columns.
// ---------------------------------------------------------------------------
__global__ __launch_bounds__(256) void gemm_a_bt_bias_lds(
    const float* __restrict__ A, const float* __restrict__ Wt,
    const float* __restrict__ bias, float* __restrict__ C,
    int N, int K, int lda, int ldb, int ldc) {
  __shared__ float As[2][BM * LDST];
  __shared__ float Bs[2][BN * LDST];

  const int tid  = threadIdx.x;
  const int wave = tid >> 5;
  const int lane = tid & 31;
  const int half = lane >> 4;   // A/B lane-half split: K={0,1} vs K={2,3}
  const int l16  = lane & 15;
  const int m0 = blockIdx.y * BM;
  const int n0 = blockIdx.x * BN;

  // Per-thread staging map: 16B chunks. A tile = 16KB -> 4 chunks/thread,
  // B tile = 8KB -> 2 chunks/thread.
  int a_row[4], a_col[4];
#pragma unroll
  for (int i = 0; i < 4; ++i) {
    int idx = i * 256 + tid;        // chunk index over 128x32 tile
    a_row[i] = idx >> 3;            // 0..127
    a_col[i] = (idx & 7) * 4;       // 0..28, 4-float chunks
  }
  int b_row[2], b_col[2], b_grow[2];
#pragma unroll
  for (int i = 0; i < 2; ++i) {
    int idx = i * 256 + tid;        // chunk index over 64x32 tile
    b_row[i] = idx >> 3;            // 0..63
    b_col[i] = (idx & 7) * 4;
    int n = n0 + b_row[i];
    b_grow[i] = n < N ? n : N - 1;  // clamp: never read OOB rows of Wt
  }

  v8f acc[4] = {v8f{}, v8f{}, v8f{}, v8f{}};

  // WMMA fragment compute over one staged BKx(BM,BN) tile.
  auto compute = [&](int buf) {
    const float* as = &As[buf][(wave * 16 + l16) * LDST + 2 * half];
    const float* bs = &Bs[buf][l16 * LDST + 2 * half];
#pragma unroll
    for (int kk = 0; kk < BK; kk += 4) {
      v2f a = *(const v2f*)(as + kk);
#pragma unroll
      for (int t = 0; t < 4; ++t) {
        v2f b = *(const v2f*)(bs + t * 16 * LDST + kk);
        acc[t] = wmma_f32(a, b, acc[t]);
      }
    }
  };

  const int NT = K / BK;

#if HAVE_ASYNC_LDS
  // Builtin signature (from clang diagnostic): arg0 = addrspace(1) int4*
  // (global source), arg1 = addrspace(3) int4* (LDS dest), imm offset, cpol.
  typedef __attribute__((address_space(1))) v4i* gv4p;
  typedef __attribute__((address_space(3))) v4i* lv4p;
  auto issue = [&](int k0, int buf) {
#pragma unroll
    for (int i = 0; i < 4; ++i) {
      gv4p g = (gv4p)(A + (size_t)(m0 + a_row[i]) * lda + k0 + a_col[i]);
      lv4p l = (lv4p)&As[buf][a_row[i] * LDST + a_col[i]];
      __builtin_amdgcn_global_load_async_to_lds_b128(g, l, 0, 0);
    }
#pragma unroll
    for (int i = 0; i < 2; ++i) {
      gv4p g = (gv4p)(Wt + (size_t)b_grow[i] * ldb + k0 + b_col[i]);
      lv4p l = (lv4p)&Bs[buf][b_row[i] * LDST + b_col[i]];
      __builtin_amdgcn_global_load_async_to_lds_b128(g, l, 0, 0);
    }
  };

  issue(0, 0);
  for (int t = 0; t < NT; ++t) {
    if (t + 1 < NT) {
      issue((t + 1) * BK, (t + 1) & 1);  // prefetch next tile
      wait_async_le6();                  // tile t's 6 copies complete
    } else {
      wait_async_le0();
    }
    __syncthreads();                     // publish LDS tile t to all waves
    compute(t & 1);
    __syncthreads();                     // safe to overwrite buf (t&1)
  }
#else
  // Fallback: global -> registers -> ds_store_b128, same double buffering.
  v4f ra[4], rb[2];
  auto gload = [&](int k0) {
#pragma unroll
    for (int i = 0; i < 4; ++i)
      ra[i] = *(const v4f*)(A + (size_t)(m0 + a_row[i]) * lda + k0 + a_col[i]);
#pragma unroll
    for (int i = 0; i < 2; ++i)
      rb[i] = *(const v4f*)(Wt + (size_t)b_grow[i] * ldb + k0 + b_col[i]);
  };
  auto lstore = [&](int buf) {
#pragma unroll
    for (int i = 0; i < 4; ++i)
      *(v4f*)&As[buf][a_row[i] * LDST + a_col[i]] = ra[i];
#pragma unroll
    for (int i = 0; i < 2; ++i)
      *(v4f*)&Bs[buf][b_row[i] * LDST + b_col[i]] = rb[i];
  };

  gload(0);
  lstore(0);
  __syncthreads();
  for (int t = 0; t < NT; ++t) {
    if (t + 1 < NT) gload((t + 1) * BK);  // global loads overlap compute
    compute(t & 1);
    if (t + 1 < NT) lstore((t + 1) & 1);
    __syncthreads();
  }
#endif

  // Epilogue: C/D layout -> VGPR r holds row (r + 8*half), col l16.
  const int rowbase = m0 + wave * 16 + 8 * half;
#pragma unroll
  for (int t = 0; t < 4; ++t) {
    int col = n0 + t * 16 + l16;
    if (col < N) {
      float bv = bias[col];
#pragma unroll
      for (int r = 0; r < 8; ++r) {
        C[(size_t)(rowbase + r) * ldc + col] = acc[t][r] + bv;
      }
    }
  }
}

// ---------------------------------------------------------------------------
// S[bh] = alpha * P_bh^T @ P_bh   (64x64 Gram matrix per batch*head).
// grid = (4 m-tiles, BS*NHEADS), block = 32.  (~0.5 GFLOP total; latency ok)
// ---------------------------------------------------------------------------
__global__ __launch_bounds__(32) void gram_ptp(
    const float* __restrict__ Pall, float* __restrict__ S, float alpha) {
  const int lane = threadIdx.x;
  const int half = lane >> 4;
  const int l16  = lane & 15;
  const int bh = blockIdx.y;
  const int b = bh >> 4;
  const int h = bh & 15;
  const int m0 = blockIdx.x * 16;

  const float* P = Pall + (size_t)b * SEQ * EMBED + h * HDIM;
  float* Sout = S + (size_t)bh * HDIM * HDIM;

  v8f acc[4] = {v8f{}, v8f{}, v8f{}, v8f{}};
  for (int k = 0; k < SEQ; k += 4) {
    const int kr = k + 2 * half;
    v2f a;                         // A[m][k] = P[k][m] (transposed access)
    a.x = P[(size_t)kr * EMBED + m0 + l16];
    a.y = P[(size_t)(kr + 1) * EMBED + m0 + l16];
#pragma unroll
    for (int t = 0; t < 4; ++t) {
      const int n = t * 16 + l16;
      v2f bb;                      // B[k][n] = P[k][n]
      bb.x = P[(size_t)kr * EMBED + n];
      bb.y = P[(size_t)(kr + 1) * EMBED + n];
      acc[t] = wmma_f32(a, bb, acc[t]);
    }
  }

  const int rowbase = m0 + 8 * half;
#pragma unroll
  for (int t = 0; t < 4; ++t) {
    const int n = t * 16 + l16;
#pragma unroll
    for (int r = 0; r < 8; ++r) {
      Sout[(rowbase + r) * HDIM + n] = alpha * acc[t][r];
    }
  }
}

// ---------------------------------------------------------------------------
// V_bh[seq,64] = P_bh[seq,64] @ S_bh[64,64], into [b, s, h*64+d] layout.
// grid = (SEQ/16, BS*NHEADS), block = 32.
// ---------------------------------------------------------------------------
__global__ __launch_bounds__(32) void apply_gram(
    const float* __restrict__ Pall, const float* __restrict__ S,
    float* __restrict__ Vall) {
  const int lane = threadIdx.x;
  const int half = lane >> 4;
  const int l16  = lane & 15;
  const int bh = blockIdx.y;
  const int b = bh >> 4;
  const int h = bh & 15;
  const int m0 = blockIdx.x * 16;

  const float* P = Pall + (size_t)b * SEQ * EMBED + h * HDIM;
  const float* Sh = S + (size_t)bh * HDIM * HDIM;
  float* V = Vall + (size_t)b * SEQ * EMBED + h * HDIM;

  const float* ap = P + (size_t)(m0 + l16) * EMBED + 2 * half;

  v8f acc[4] = {v8f{}, v8f{}, v8f{}, v8f{}};
#pragma unroll
  for (int k = 0; k < HDIM; k += 4) {
    v2f a = *(const v2f*)(ap + k);
    const int kr = k + 2 * half;
#pragma unroll
    for (int t = 0; t < 4; ++t) {
      const int n = t * 16 + l16;
      v2f bb;                      // B[k][n] = S[k][n] (row-major)
      bb.x = Sh[kr * HDIM + n];
      bb.y = Sh[(kr + 1) * HDIM + n];
      acc[t] = wmma_f32(a, bb, acc[t]);
    }
  }

  const int rowbase = m0 + 8 * half;
#pragma unroll
  for (int t = 0; t < 4; ++t) {
    const int n = t * 16 + l16;
#pragma unroll
    for (int r = 0; r < 8; ++r) {
      V[(size_t)(rowbase + r) * EMBED + n] = acc[t][r];
    }
  }
}

extern "C" void kernel_launch(void* const* d_in, const int* in_sizes, int n_in,
                              void* d_out, int out_size, void* d_ws, size_t ws_size,
                              hipStream_t stream) {
  const float* x     = (const float*)d_in[0];  // [2, 2048, 1024]
  const float* W_in  = (const float*)d_in[1];  // [1024, 1024]
  const float* b_in  = (const float*)d_in[2];  // [1024]
  const float* W_out = (const float*)d_in[3];  // [1000, 1024]
  const float* b_out = (const float*)d_in[4];  // [1000]
  float* out = (float*)d_out;                  // [2, 2048, 1000]

  const int M = BS * SEQ;                      // 4096
  float* p    = (float*)d_ws;                  // M * EMBED floats   (16 MB)
  float* S    = p + (size_t)M * EMBED;         // 32 * 64 * 64       (0.5 MB)
  float* vals = S + (size_t)BS * NHEADS * HDIM * HDIM;  // M * EMBED (16 MB)

  const float scale = 1.0f / 32.0f;            // 1/sqrt(EMBED)

  // 1) p = x @ W_in^T + b_in
  gemm_a_bt_bias_lds<<<dim3(EMBED / BN, M / BM), 256, 0, stream>>>(
      x, W_in, b_in, p, EMBED, EMBED, EMBED, EMBED, EMBED);

  // 2) S_bh = scale * p_bh^T @ p_bh
  gram_ptp<<<dim3(HDIM / 16, BS * NHEADS), 32, 0, stream>>>(p, S, scale);

  // 3) vals_bh = p_bh @ S_bh
  apply_gram<<<dim3(SEQ / 16, BS * NHEADS), 32, 0, stream>>>(p, S, vals);

  // 4) out = vals @ W_out^T + b_out   (N=1000, clamped/guarded)
  gemm_a_bt_bias_lds<<<dim3((NCLASSES + BN - 1) / BN, M / BM), 256, 0, stream>>>(
      vals, W_out, b_out, out, NCLASSES, EMBED, EMBED, EMBED, NCLASSES);
}